// EMAVectorQuantizer_73117523247500
// MI455X (gfx1250) — compile-verified
//
#include <hip/hip_runtime.h>
#include <math.h>

typedef float v2f __attribute__((ext_vector_type(2)));
typedef float v8f __attribute__((ext_vector_type(8)));

#define NE    4096      // number of codes (K in sinkhorn)
#define MPTS  16384     // number of latents (N in sinkhorn)
#define EDIM  256
#define DECAY 0.99f
#define LEPS  1e-5

// ---------------- workspace layout (bytes); needs ws_size >= ~274 MB ----------------
static constexpr size_t OFF_D      = 0;                                  // 16384x4096 f32
static constexpr size_t OFF_A2     = (size_t)MPTS * NE * 4;              // 16384 f32
static constexpr size_t OFF_B2     = OFF_A2 + (size_t)MPTS * 4;          // 4096 f32
static constexpr size_t OFF_KEYS   = OFF_B2 + (size_t)NE * 4;            // 2 u32 (max,min keys)
static constexpr size_t OFF_SCAL   = OFF_KEYS + 256;                     // 2 f32 (middle, 1/amp)
static constexpr size_t OFF_V      = OFF_SCAL + 256;                     // 16384 f64
static constexpr size_t OFF_ROWDOT = OFF_V + (size_t)MPTS * 8;           // 4096 f64
static constexpr size_t OFF_LOGU   = OFF_ROWDOT + (size_t)NE * 8;        // 4096 f64
static constexpr size_t OFF_U      = OFF_LOGU + (size_t)NE * 8;          // 4096 f64
static constexpr size_t OFF_IDX    = OFF_U + (size_t)NE * 8;             // 16384 i32
static constexpr size_t OFF_CNT    = OFF_IDX + (size_t)MPTS * 4;         // 4096 f32
static constexpr size_t OFF_ESUM   = OFF_CNT + (size_t)NE * 4;           // 4096x256 f32
static constexpr size_t OFF_LOSS   = OFF_ESUM + (size_t)NE * EDIM * 4;   // f64
static constexpr size_t OFF_NSUM   = OFF_LOSS + 8;                       // f64

// ---------------- output layout (floats, reference return order) ----------------
static constexpr size_t O_XQ   = 0;
static constexpr size_t O_LOSS = O_XQ + (size_t)MPTS * EDIM;
static constexpr size_t O_IDX  = O_LOSS + 1;
static constexpr size_t O_NEMB = O_IDX + MPTS;
static constexpr size_t O_NCLU = O_NEMB + (size_t)NE * EDIM;
static constexpr size_t O_NAVG = O_NCLU + NE;

// monotonic float <-> unsigned order-preserving map (for atomic min/max)
__device__ __forceinline__ unsigned flip_f(float f) {
  unsigned u = __float_as_uint(f);
  return (u & 0x80000000u) ? ~u : (u | 0x80000000u);
}
__device__ __forceinline__ float unflip_f(unsigned u) {
  return (u & 0x80000000u) ? __uint_as_float(u ^ 0x80000000u) : __uint_as_float(~u);
}

// ---- CDNA5 async Global->LDS copy (ASYNCcnt-tracked), GV addressing mode ----
__device__ __forceinline__ void async_ld_b128(unsigned lds_off, const void* gaddr) {
  asm volatile("global_load_async_to_lds_b128 %0, %1, off"
               :: "v"(lds_off), "v"((unsigned long long)gaddr)
               : "memory");
}
__device__ __forceinline__ void wait_async0() {
  asm volatile("s_wait_asynccnt 0x0" ::: "memory");
}

// -------- per-row sum of squares (wave per row) --------
__global__ void k_rowsq(const float* __restrict__ m, float* __restrict__ out, int rows) {
  const int w = (int)((blockIdx.x * blockDim.x + threadIdx.x) >> 5);
  const int lane = threadIdx.x & 31;
  if (w >= rows) return;
  const float* r = m + (size_t)w * EDIM;
  float s = 0.0f;
  for (int c = lane; c < EDIM; c += 32) { float v = r[c]; s += v * v; }
  #pragma unroll
  for (int o = 16; o; o >>= 1) s += __shfl_xor(s, o, 32);
  if (lane == 0) out[w] = s;
}

// -------- init workspace state --------
__global__ void k_init(unsigned* keys, double* v, double* rowdot,
                       float* cnt, float* esum, double* loss, double* nsum) {
  const size_t g = (size_t)blockIdx.x * blockDim.x + threadIdx.x;
  if (g == 0) { keys[0] = 0u; keys[1] = 0xFFFFFFFFu; *loss = 0.0; *nsum = 0.0; }
  if (g < MPTS) v[g] = 1.0;
  if (g < NE)   { rowdot[g] = 0.0; cnt[g] = 0.0f; }
  if (g < (size_t)NE * EDIM) esum[g] = 0.0f;
}

// -------- fused distance GEMM: D = a2 + b2 - 2*A@B^T, with global min/max --------
// block 256 thr = 8 waves, tile 64x128, wave tile 16x64, WMMA f32 16x16x4.
// K staged 32-wide, double-buffered via async Global->LDS copies so tile t+1's
// copy overlaps tile t's 32-WMMA stream.
__launch_bounds__(256)
__global__ void k_gemm(const float* __restrict__ A, const float* __restrict__ B,
                       const float* __restrict__ a2, const float* __restrict__ b2,
                       float* __restrict__ D, unsigned* __restrict__ keys)
{
  __shared__ float As[2][64 * 36];    // stride 36 floats: 16B rows, conflict-free frags
  __shared__ float Bs[2][128 * 36];
  const int tid  = threadIdx.x;
  const int lane = tid & 31;
  const int wave = tid >> 5;
  const int wm = wave >> 1, wn = wave & 1;       // 4x2 wave grid
  const int hi = lane >> 4, l16 = lane & 15;
  const int bm = blockIdx.x * 64, bn = blockIdx.y * 128;
  const int lr = tid >> 3;                        // staging row 0..31
  const int lc = (tid & 7) * 4;                   // staging col (float4)

  v8f acc[4];
  #pragma unroll
  for (int j = 0; j < 4; ++j)
    #pragma unroll
    for (int r = 0; r < 8; ++r) acc[j][r] = 0.0f;

  auto stage = [&](int k0, int buf) {
    #pragma unroll
    for (int r = 0; r < 2; ++r) {
      const int row = lr + r * 32;
      async_ld_b128((unsigned)(uintptr_t)&As[buf][row * 36 + lc],
                    A + (size_t)(bm + row) * EDIM + k0 + lc);
    }
    #pragma unroll
    for (int r = 0; r < 4; ++r) {
      const int row = lr + r * 32;
      async_ld_b128((unsigned)(uintptr_t)&Bs[buf][row * 36 + lc],
                    B + (size_t)(bn + row) * EDIM + k0 + lc);
    }
  };

  stage(0, 0);
  wait_async0();
  __syncthreads();

  for (int t = 0; t < EDIM / 32; ++t) {
    const int buf = t & 1;
    if (t + 1 < EDIM / 32) stage((t + 1) * 32, buf ^ 1);   // overlap copy with WMMA

    const float* as = As[buf];
    const float* bs = Bs[buf];
    #pragma unroll
    for (int ks = 0; ks < 32; ks += 4) {
      // A frag: lane holds row l16, k-pair (ks+2*hi, ks+2*hi+1)
      const v2f a = *(const v2f*)&as[(wm * 16 + l16) * 36 + ks + 2 * hi];
      #pragma unroll
      for (int j = 0; j < 4; ++j) {
        // B frag: lane holds col l16 of B^T => row (wn*64+j*16+l16) of emb
        const v2f b = *(const v2f*)&bs[(wn * 64 + j * 16 + l16) * 36 + ks + 2 * hi];
        acc[j] = __builtin_amdgcn_wmma_f32_16x16x4_f32(
            false, a, false, b, (short)0, acc[j], false, false);
      }
    }
    wait_async0();     // tile t+1 resident
    __syncthreads();   // all waves done reading buf before it is overwritten next iter
  }

  float lmin = 3.4e38f, lmax = -3.4e38f;
  #pragma unroll
  for (int j = 0; j < 4; ++j) {
    const int gn = bn + wn * 64 + j * 16 + l16;
    const float bb = b2[gn];
    #pragma unroll
    for (int r = 0; r < 8; ++r) {
      const int gm = bm + wm * 16 + r + 8 * hi;   // C/D layout: m = r + 8*hi, n = lane&15
      const float dv = a2[gm] + bb - 2.0f * acc[j][r];
      D[(size_t)gm * NE + gn] = dv;
      lmin = fminf(lmin, dv);
      lmax = fmaxf(lmax, dv);
    }
  }
  #pragma unroll
  for (int o = 16; o; o >>= 1) {
    lmax = fmaxf(lmax, __shfl_xor(lmax, o, 32));
    lmin = fminf(lmin, __shfl_xor(lmin, o, 32));
  }
  if (lane == 0) {
    atomicMax(&keys[0], flip_f(lmax));
    atomicMin(&keys[1], flip_f(lmin));
  }
}

// -------- derive middle & 1/amplitude --------
__global__ void k_scal(const unsigned* __restrict__ keys, float* __restrict__ scal) {
  if (threadIdx.x == 0) {
    const float dmax = unflip_f(keys[0]);
    const float dmin = unflip_f(keys[1]);
    const float mid = (dmax + dmin) * 0.5f;
    const float amp = dmax - mid + 1e-5f;
    scal[0] = mid;
    scal[1] = 1.0f / amp;
  }
}

// -------- sinkhorn pass A: rowdot_k += sum_n E[k,n]*v[n]  (E recomputed from D) --------
__global__ void k_rowdot(const float* __restrict__ D, const double* __restrict__ v,
                         const float* __restrict__ scal, double* __restrict__ rowdot) {
  const int k  = blockIdx.y * 256 + threadIdx.x;  // gridDim.y = NE/256
  const int n0 = blockIdx.x * 128;                // gridDim.x = MPTS/128
  const float mid = scal[0], invamp = scal[1];
  double acc = 0.0;
  for (int i = 0; i < 128; ++i) {
    const int n = n0 + i;
    const float dc = (D[(size_t)n * NE + k] - mid) * invamp;      // f32 like reference
    acc += exp(-(double)dc * (1.0 / 0.003)) * v[n];
  }
  atomicAdd(&rowdot[k], acc);
}

// -------- u_k = 1/(K*rowdot_k); also log u; re-zero rowdot for next iteration --------
__global__ void k_finalize_u(double* __restrict__ rowdot, double* __restrict__ u,
                             double* __restrict__ logu) {
  const int k = blockIdx.x * 256 + threadIdx.x;
  const double uk = 1.0 / ((double)NE * rowdot[k]);
  u[k] = uk;
  logu[k] = log(uk);
  rowdot[k] = 0.0;
}

// -------- sinkhorn pass B (wave per point): v_n = 1/(N * sum_k u_k E[k,n]) --------
__global__ void k_coldot(const float* __restrict__ D, const double* __restrict__ u,
                         const float* __restrict__ scal, double* __restrict__ v) {
  const int n = (int)((blockIdx.x * blockDim.x + threadIdx.x) >> 5);
  const int lane = threadIdx.x & 31;
  if (n >= MPTS) return;
  const float mid = scal[0], invamp = scal[1];
  const float* row = D + (size_t)n * NE;
  double acc = 0.0;
  for (int k = lane; k < NE; k += 32) {
    const float dc = (row[k] - mid) * invamp;
    acc += u[k] * exp(-(double)dc * (1.0 / 0.003));
  }
  #pragma unroll
  for (int o = 16; o; o >>= 1) acc += __shfl_xor(acc, o, 32);
  if (lane == 0) v[n] = 1.0 / ((double)MPTS * acc);
}

// -------- argmax_k of log u_k - d_c/eps (wave per point) --------
__global__ void k_argmax(const float* __restrict__ D, const double* __restrict__ logu,
                         const float* __restrict__ scal, int* __restrict__ idx) {
  const int n = (int)((blockIdx.x * blockDim.x + threadIdx.x) >> 5);
  const int lane = threadIdx.x & 31;
  if (n >= MPTS) return;
  const float mid = scal[0], invamp = scal[1];
  const float* row = D + (size_t)n * NE;
  double best = -1.0e300;
  int bi = 0;
  for (int k = lane; k < NE; k += 32) {
    const float dc = (row[k] - mid) * invamp;
    const double s = logu[k] - (double)dc * (1.0 / 0.003);
    if (s > best) { best = s; bi = k; }
  }
  #pragma unroll
  for (int o = 16; o; o >>= 1) {
    const double ob = __shfl_xor(best, o, 32);
    const int    oi = __shfl_xor(bi, o, 32);
    if (ob > best || (ob == best && oi < bi)) { best = ob; bi = oi; }
  }
  if (lane == 0) idx[n] = bi;
}

// -------- scatter: embedding_sum += onehot^T @ latent ; counts --------
__global__ void k_scatter(const float* __restrict__ latent, const int* __restrict__ idx,
                          float* __restrict__ esum, float* __restrict__ cnt) {
  const int n = blockIdx.x, c = threadIdx.x;
  const int k = idx[n];
  atomicAdd(&esum[(size_t)k * EDIM + c], latent[(size_t)n * EDIM + c]);
  if (c == 0) atomicAdd(&cnt[k], 1.0f);
}

// -------- x_q gather + commitment-loss partial --------
__global__ void k_xq_loss(const float* __restrict__ x, const float* __restrict__ emb,
                          const int* __restrict__ idx, float* __restrict__ out,
                          double* __restrict__ loss) {
  const int n = blockIdx.x, c = threadIdx.x;
  const int k = idx[n];
  const float q  = emb[(size_t)k * EDIM + c];
  const float xv = x[(size_t)n * EDIM + c];
  out[O_XQ + (size_t)n * EDIM + c] = q;     // x + sg(x_q - x) == x_q in value
  const float df = q - xv;
  float s = df * df;
  __shared__ float sw[8];
  #pragma unroll
  for (int o = 16; o; o >>= 1) s += __shfl_xor(s, o, 32);
  if ((threadIdx.x & 31) == 0) sw[threadIdx.x >> 5] = s;
  __syncthreads();
  if (threadIdx.x == 0) {
    float t = 0.0f;
    #pragma unroll
    for (int i = 0; i < 8; ++i) t += sw[i];
    atomicAdd(loss, (double)t);
  }
}

// -------- new_cluster --------
__global__ void k_newclu(const float* __restrict__ cs, const float* __restrict__ cnt,
                         float* __restrict__ out) {
  const int k = blockIdx.x * 256 + threadIdx.x;
  out[O_NCLU + k] = cs[k] * DECAY + cnt[k] * (1.0f - DECAY);
}

// -------- n = sum(new_cluster); finalize loss --------
__global__ void k_nsum(const float* __restrict__ nclu, const double* __restrict__ lossacc,
                       double* __restrict__ nsum, float* __restrict__ out) {
  __shared__ double sd[256];
  double s = 0.0;
  for (int i = threadIdx.x; i < NE; i += 256) s += (double)nclu[i];
  sd[threadIdx.x] = s;
  __syncthreads();
  for (int st = 128; st; st >>= 1) {
    if (threadIdx.x < st) sd[threadIdx.x] += sd[threadIdx.x + st];
    __syncthreads();
  }
  if (threadIdx.x == 0) {
    *nsum = sd[0];
    out[O_LOSS] = (float)(0.25 * (*lossacc) / (double)((size_t)MPTS * EDIM));
  }
}

// -------- indices to float output --------
__global__ void k_idxf(const int* __restrict__ idx, float* __restrict__ out) {
  const int n = blockIdx.x * 256 + threadIdx.x;
  out[O_IDX + n] = (float)idx[n];
}

// -------- EMA update + Laplace smoothing + dead-code replacement --------
__global__ void k_update(const float* __restrict__ eavg, const float* __restrict__ esum,
                         const float* __restrict__ latent, const double* __restrict__ nsum,
                         float* __restrict__ out) {
  const int k = blockIdx.x, c = threadIdx.x;
  const size_t o = (size_t)k * EDIM + c;
  const float avg = eavg[o] * DECAY + esum[o] * (1.0f - DECAY);
  out[O_NAVG + o] = avg;
  const float ncl = out[O_NCLU + k];
  const double ns = *nsum;
  const float smoothed = (float)((((double)ncl + LEPS) / (ns + (double)NE * LEPS)) * ns);
  const float en = avg / smoothed;
  const float usage = (ncl >= 1.0f) ? 1.0f : 0.0f;
  // deterministic stand-in for jax.random.permutation(key(42)) (bijection over 16384 rows)
  const unsigned p = (unsigned)(k * 2654435761u) & (MPTS - 1);
  const float tmp = latent[(size_t)p * EDIM + c];
  out[O_NEMB + o] = usage * en + (1.0f - usage) * tmp;
}

extern "C" void kernel_launch(void* const* d_in, const int* in_sizes, int n_in,
                              void* d_out, int out_size, void* d_ws, size_t ws_size,
                              hipStream_t stream) {
  (void)in_sizes; (void)n_in; (void)out_size; (void)ws_size;
  const float* x    = (const float*)d_in[0];   // 32*512*256 -> latent 16384x256
  const float* emb  = (const float*)d_in[1];   // 4096x256
  const float* eavg = (const float*)d_in[2];   // 4096x256
  const float* cs   = (const float*)d_in[3];   // 4096
  float* out = (float*)d_out;
  char* ws = (char*)d_ws;

  float*    D       = (float*)(ws + OFF_D);
  float*    a2      = (float*)(ws + OFF_A2);
  float*    b2      = (float*)(ws + OFF_B2);
  unsigned* keys    = (unsigned*)(ws + OFF_KEYS);
  float*    scal    = (float*)(ws + OFF_SCAL);
  double*   v       = (double*)(ws + OFF_V);
  double*   rowdot  = (double*)(ws + OFF_ROWDOT);
  double*   logu    = (double*)(ws + OFF_LOGU);
  double*   u       = (double*)(ws + OFF_U);
  int*      idx     = (int*)(ws + OFF_IDX);
  float*    cnt     = (float*)(ws + OFF_CNT);
  float*    esum    = (float*)(ws + OFF_ESUM);
  double*   lossacc = (double*)(ws + OFF_LOSS);
  double*   nsum    = (double*)(ws + OFF_NSUM);

  k_rowsq<<<MPTS / 8, 256, 0, stream>>>(x, a2, MPTS);
  k_rowsq<<<NE / 8, 256, 0, stream>>>(emb, b2, NE);
  k_init<<<(NE * EDIM) / 256, 256, 0, stream>>>(keys, v, rowdot, cnt, esum, lossacc, nsum);
  k_gemm<<<dim3(MPTS / 64, NE / 128), 256, 0, stream>>>(x, emb, a2, b2, D, keys);
  k_scal<<<1, 32, 0, stream>>>(keys, scal);
  for (int it = 0; it < 10; ++it) {
    k_rowdot<<<dim3(MPTS / 128, NE / 256), 256, 0, stream>>>(D, v, scal, rowdot);
    k_finalize_u<<<NE / 256, 256, 0, stream>>>(rowdot, u, logu);
    k_coldot<<<MPTS / 8, 256, 0, stream>>>(D, u, scal, v);
  }
  k_argmax<<<MPTS / 8, 256, 0, stream>>>(D, logu, scal, idx);
  k_scatter<<<MPTS, 256, 0, stream>>>(x, idx, esum, cnt);
  k_xq_loss<<<MPTS, 256, 0, stream>>>(x, emb, idx, out, lossacc);
  k_newclu<<<NE / 256, 256, 0, stream>>>(cs, cnt, out);
  k_nsum<<<1, 256, 0, stream>>>(out + O_NCLU, lossacc, nsum, out);
  k_idxf<<<MPTS / 256, 256, 0, stream>>>(idx, out);
  k_update<<<NE, 256, 0, stream>>>(eavg, esum, x, nsum, out);
}